// LightGCNBaseline_26989574488331
// MI455X (gfx1250) — compile-verified
//
#include <hip/hip_runtime.h>

// LightGCN propagation on MI455X (gfx1250).
//
// Roofline: 1.9 GFLOP total vs ~76.8MB node state that fits in the 192MB L2.
// => L2-atomic / L2-bandwidth bound scatter SpMM; WMMA used for the exact
// dense tile accumulation acc += x (identity-A chained V_WMMA_F32_16X16X4_F32,
// exact in f32, bandwidth-bound so the matrix pipe costs nothing).

#define NUM_USERS 100000
#define NUM_ITEMS 50000
#define N_NODES   150000
#define DIM       128
#define N_LAYERS  3

typedef __attribute__((ext_vector_type(2))) float v2f;
typedef __attribute__((ext_vector_type(8))) float v8f;

// Guaranteed-native fp32 atomic add, device scope (cross-WGP coherent in L2).
__device__ __forceinline__ void atomic_add_f32(float* p, float v) {
  asm volatile("global_atomic_add_f32 %0, %1, off scope:SCOPE_DEV"
               :: "v"(p), "v"(v) : "memory");
}

// out = x = concat(user_emb, item_emb), float4-vectorized (128-dim rows => aligned).
__global__ __launch_bounds__(256)
void k_init(const float4* __restrict__ ue, const float4* __restrict__ ie,
            float4* __restrict__ x, float4* __restrict__ out) {
  long long i = (long long)blockIdx.x * blockDim.x + threadIdx.x;
  const long long nu = (long long)NUM_USERS * DIM / 4;
  const long long nt = (long long)N_NODES * DIM / 4;
  if (i >= nt) return;
  float4 v = (i < nu) ? ue[i] : ie[i - nu];
  x[i]   = v;
  out[i] = v;
}

__global__ __launch_bounds__(256)
void k_zero(float4* __restrict__ p, long long n4) {
  long long i = (long long)blockIdx.x * blockDim.x + threadIdx.x;
  if (i < n4) p[i] = make_float4(0.f, 0.f, 0.f, 0.f);
}

// COO scatter SpMM: one wave32 per edge. 32 lanes x float4 = 128 dims.
// Gather x[col] is a contiguous 512B wave load (L2-resident), scatter is
// 4 native f32 atomics per lane into x_new[row].
__global__ __launch_bounds__(256)
void k_scatter(const int* __restrict__ rows, const int* __restrict__ cols,
               const float* __restrict__ vals, const float* __restrict__ x,
               float* __restrict__ xn, int nnz) {
  int lane = threadIdx.x & 31;
  int e = blockIdx.x * (blockDim.x >> 5) + (threadIdx.x >> 5);
  if (e >= nnz) return;
  int r = rows[e];
  int c = cols[e];
  float v = vals[e];
  const float4* src = (const float4*)(x + (long long)c * DIM);
  float4 d = src[lane];
  float* dst = xn + (long long)r * DIM + lane * 4;
  atomic_add_f32(dst + 0, v * d.x);
  atomic_add_f32(dst + 1, v * d.y);
  atomic_add_f32(dst + 2, v * d.z);
  atomic_add_f32(dst + 3, v * d.w);
}

// acc(out) += xn, one 16x16 f32 tile per wave via 4 chained WMMAs:
//   D = A_s x B_s + C,  A_s[m][k] = (m == 4s+k)  (exact identity in f32)
//   B_s = rows [4s, 4s+4) of the x tile  =>  D = x_tile + acc_tile.
// A layout (ISA 7.12.2, 32-bit 16x4): lane l holds A[l%16][k0a], A[l%16][k0a+1],
//   k0a = 2*(l>=16). B follows the C/D half-wave row-split pattern
//   (VGPR j: rows {j, j+2}). C/D: VGPR j = rows {j, j+8}, col = lane%16.
// EXEC is all-1s: 75000 tiles / 8 waves = 9375 blocks exactly, no divergence.
__global__ __launch_bounds__(256)
void k_accum_wmma(const float* __restrict__ xn, float* __restrict__ out) {
  int lane = threadIdx.x & 31;
  int wv   = threadIdx.x >> 5;
  int t    = blockIdx.x * 8 + wv;     // tile id, 75000 total
  int mt   = t >> 3;                  // node tile (16 rows)
  int nt   = t & 7;                   // dim tile (16 cols)
  int half = lane >> 4;               // 0: rows j / 1: rows j+8 (C layout)
  int lc   = lane & 15;               // column within tile
  long long base = (long long)mt * 16 * DIM + nt * 16 + lc;

  v8f c;
  #pragma unroll
  for (int j = 0; j < 8; ++j)
    c[j] = out[base + (long long)(j + 8 * half) * DIM];

  #pragma unroll
  for (int s = 0; s < 4; ++s) {
    int k0 = 4 * s + 2 * half;
    v2f a, b;
    a.x = (lc == k0)     ? 1.0f : 0.0f;   // shifted identity (exact)
    a.y = (lc == k0 + 1) ? 1.0f : 0.0f;
    b.x = xn[base + (long long)(k0)     * DIM];
    b.y = xn[base + (long long)(k0 + 1) * DIM];
    c = __builtin_amdgcn_wmma_f32_16x16x4_f32(
        /*neg_a=*/false, a, /*neg_b=*/false, b,
        /*c_mod=*/(short)0, c, /*reuse_a=*/false, /*reuse_b=*/false);
  }

  #pragma unroll
  for (int j = 0; j < 8; ++j)
    out[base + (long long)(j + 8 * half) * DIM] = c[j];
}

__global__ __launch_bounds__(256)
void k_scale(float4* __restrict__ p, long long n4) {
  long long i = (long long)blockIdx.x * blockDim.x + threadIdx.x;
  if (i < n4) {
    float4 v = p[i];
    v.x *= 0.25f; v.y *= 0.25f; v.z *= 0.25f; v.w *= 0.25f;
    p[i] = v;
  }
}

extern "C" void kernel_launch(void* const* d_in, const int* in_sizes, int n_in,
                              void* d_out, int out_size, void* d_ws, size_t ws_size,
                              hipStream_t stream) {
  const float* ue   = (const float*)d_in[0];   // user_emb  [100000,128] f32
  const float* ie   = (const float*)d_in[1];   // item_emb  [ 50000,128] f32
  const int*   rows = (const int*)d_in[2];     // [NNZ] i32
  const int*   cols = (const int*)d_in[3];     // [NNZ] i32
  const float* vals = (const float*)d_in[4];   // [NNZ] f32
  int nnz = in_sizes[2];

  float* out = (float*)d_out;                          // acc, 150000x128 f32
  float* x   = (float*)d_ws;                           // 76.8 MB
  float* xn  = (float*)d_ws + (size_t)N_NODES * DIM;   // 76.8 MB

  const long long total4 = (long long)N_NODES * DIM / 4;   // 4.8M float4
  const int B = 256;
  const int gElem    = (int)((total4 + B - 1) / B);                  // 18750
  const int gScatter = (nnz + 7) / 8;                                // 8 waves/block
  const int gTiles   = (N_NODES / 16) * (DIM / 16) / 8;              // 9375

  k_init<<<gElem, B, 0, stream>>>((const float4*)ue, (const float4*)ie,
                                  (float4*)x, (float4*)out);

  for (int l = 0; l < N_LAYERS; ++l) {
    k_zero<<<gElem, B, 0, stream>>>((float4*)xn, total4);
    k_scatter<<<gScatter, B, 0, stream>>>(rows, cols, vals, x, xn, nnz);
    k_accum_wmma<<<gTiles, B, 0, stream>>>(xn, out);
    float* tmp = x; x = xn; xn = tmp;   // ping-pong
  }

  k_scale<<<gElem, B, 0, stream>>>((float4*)out, total4);
}